// PositionalEncoding_posi_72215580115573
// MI455X (gfx1250) — compile-verified
//
#include <hip/hip_runtime.h>
#include <math.h>

// Problem constants from the reference: B=16, L=4096, D=2048.
#define BB 16
#define LL 4096
#define DD 2048

// Native clang vector types (required by __builtin_nontemporal_*).
typedef float v4f __attribute__((ext_vector_type(4)));
typedef float v2f __attribute__((ext_vector_type(2)));

// TAB[rot][case][xy]; case 4 row is all-zero (np.zeros init in reference).
__device__ __constant__ int TAB[8][5][2] = {
  {{ 1, 0},{-1, 0},{ 0,-1},{ 0, 1},{0,0}},
  {{ 1, 1},{-1,-1},{ 1,-1},{-1, 1},{0,0}},
  {{ 0, 1},{ 0,-1},{ 1, 0},{-1, 0},{0,0}},
  {{-1, 1},{ 1,-1},{ 1, 1},{-1,-1},{0,0}},
  {{-1, 0},{ 1, 0},{ 0, 1},{ 0,-1},{0,0}},
  {{-1,-1},{ 1, 1},{-1, 1},{ 1,-1},{0,0}},
  {{ 0,-1},{ 0, 1},{-1, 0},{ 1, 0},{0,0}},
  {{ 1,-1},{-1, 1},{-1,-1},{ 1, 1},{0,0}},
};

// ---------------------------------------------------------------------------
// Kernel 1: parallel inclusive cumsum of the per-step (dx,dy) deltas.
// One workgroup of 1024 threads (32 wave32 waves on one WGP); each thread
// serially folds 4 consecutive actions, then a Hillis-Steele scan over the
// 1024 chunk totals in LDS, then local results are rebased. Writes v2f
// counts[L] into workspace.
// ---------------------------------------------------------------------------
__global__ __launch_bounds__(1024) void scan_counts(const int* __restrict__ action,
                                                    v2f* __restrict__ counts) {
  __shared__ int sx[1024];
  __shared__ int sy[1024];
  const int t = threadIdx.x;

  int lx[4], ly[4];
  int ax = 0, ay = 0;
#pragma unroll
  for (int i = 0; i < 4; ++i) {
    const int l  = t * 4 + i;
    const int i0 = action[3 * l + 0];
    const int i1 = action[3 * l + 1];
    const int r  = action[3 * l + 2] & 7;   // rot is in [0,8) per reference
    const int c  = (i0 == 1) ? 0 : (i0 == -1) ? 1 : (i1 == 1) ? 2 : (i1 == -1) ? 3 : 4;
    ax += TAB[r][c][0];
    ay += TAB[r][c][1];
    lx[i] = ax;                              // inclusive within chunk
    ly[i] = ay;
  }
  sx[t] = ax;
  sy[t] = ay;
  __syncthreads();

  // Hillis-Steele inclusive scan of the 1024 chunk totals.
  for (int off = 1; off < 1024; off <<= 1) {
    int vx = 0, vy = 0;
    if (t >= off) { vx = sx[t - off]; vy = sy[t - off]; }
    __syncthreads();
    sx[t] += vx;
    sy[t] += vy;
    __syncthreads();
  }

  const int ox = sx[t] - ax;                 // exclusive offset for this chunk
  const int oy = sy[t] - ay;
#pragma unroll
  for (int i = 0; i < 4; ++i) {
    v2f c;
    c.x = (float)(lx[i] + ox);
    c.y = (float)(ly[i] + oy);
    counts[t * 4 + i] = c;
  }
}

// ---------------------------------------------------------------------------
// Kernel 2: out[b,l,d] = x[b,l,d] + pe[l,d], b128-vectorized.
// One thread per (l, group of 4 channels): compute the 4 pe values ONCE
// (2 sinf + 2 cosf), then stream all 16 batch slices through b128 loads and
// stores with non-temporal hints (x is read-once, out is write-once; keep
// the 1 GiB stream out of L2 so counts/div_term stay resident).
// ---------------------------------------------------------------------------
__global__ __launch_bounds__(256) void pe_add(const v4f* __restrict__ x4,
                                              const v2f* __restrict__ counts,
                                              const v2f* __restrict__ div2,
                                              v4f* __restrict__ out4) {
  const int gid = blockIdx.x * 256 + threadIdx.x;  // 0 .. L*(D/4)-1
  const int l = gid >> 9;                          // D/4 = 512 v4f per row
  const int j = gid & 511;

  const v2f c  = counts[l];                        // (count_x, count_y) for row l
  const v2f dv = div2[j];                          // div_term[2j], div_term[2j+1]

  v4f pe;
  pe.x = sinf(c.x * dv.x);
  pe.y = cosf(c.y * dv.x);
  pe.z = sinf(c.x * dv.y);
  pe.w = cosf(c.y * dv.y);

  const size_t base   = (size_t)l * (DD / 4) + (size_t)j;
  const size_t stride = (size_t)LL * (DD / 4);     // one batch slice, in v4f

  v4f v[BB];
#pragma unroll
  for (int b = 0; b < BB; ++b) {                   // cluster the 16 b128 loads
    v[b] = __builtin_nontemporal_load(&x4[base + (size_t)b * stride]);
  }
#pragma unroll
  for (int b = 0; b < BB; ++b) {
    __builtin_nontemporal_store(v[b] + pe, &out4[base + (size_t)b * stride]);
  }
}

// ---------------------------------------------------------------------------
// Launch. Inputs per reference setup_inputs() order:
//   d_in[0] = x        float32 [16, 4096, 2048]
//   d_in[1] = action   int32   [4096, 3]
//   d_in[2] = div_term float32 [1024]
// d_out = float32 [16, 4096, 2048]; d_ws holds counts (4096 * v2f = 32 KB).
// ---------------------------------------------------------------------------
extern "C" void kernel_launch(void* const* d_in, const int* in_sizes, int n_in,
                              void* d_out, int out_size, void* d_ws, size_t ws_size,
                              hipStream_t stream) {
  const float* x        = (const float*)d_in[0];
  const int*   action   = (const int*)d_in[1];
  const float* div_term = (const float*)d_in[2];

  v2f* counts = (v2f*)d_ws;

  scan_counts<<<1, 1024, 0, stream>>>(action, counts);

  const int total4 = LL * (DD / 4);                // 2,097,152 threads
  pe_add<<<total4 / 256, 256, 0, stream>>>((const v4f*)x, counts,
                                           (const v2f*)div_term,
                                           (v4f*)d_out);
}